// GIN_60078002536566
// MI455X (gfx1250) — compile-verified
//
#include <hip/hip_runtime.h>
#include <hip/hip_bf16.h>
#include <stdint.h>

// ---------------------------------------------------------------------------
// GIN forward: 3 x ( h' = (h + segment_sum(h[src], dst)) @ W ), fp32.
// N = 10000 (multiple of 16), D = 128, layer dims 128->128->128->64.
// ---------------------------------------------------------------------------

typedef float v2f __attribute__((ext_vector_type(2)));
typedef float v8f __attribute__((ext_vector_type(8)));

// Native device-scope fp32 atomic add (GLOBAL_ATOMIC_ADD_F32, no return ->
// STOREcnt). Inline asm guarantees the native instruction, not a CAS loop.
__device__ __forceinline__ void atomic_add_f32_dev(float* p, float v) {
    asm volatile("global_atomic_add_f32 %0, %1, off scope:SCOPE_DEV"
                 :: "v"((unsigned long long)(uintptr_t)p), "v"(v)
                 : "memory");
}

// t[i] = h[i]  (vectorized; also fully initializes the poisoned workspace)
__global__ void gin_copy_f32x4(const float4* __restrict__ src,
                               float4* __restrict__ dst, int n4) {
    int i = blockIdx.x * blockDim.x + threadIdx.x;
    if (i < n4) dst[i] = src[i];
}

// acc[dst[e]][:] += h[src[e]][:]
// One wave per edge; each lane handles 4 consecutive features (float4 gather,
// 4 native fp32 atomics). Working set (5.1 MB) is L2-resident on MI455X.
__global__ void gin_scatter_add(const float* __restrict__ h,
                                const int* __restrict__ src,
                                const int* __restrict__ dst,
                                float* __restrict__ acc, int nE) {
    int t = blockIdx.x * blockDim.x + threadIdx.x;
    int e = t >> 5;          // wave-per-edge (wave32)
    int lane = t & 31;       // lane -> feature chunk
    if (e >= nE) return;
    int s = src[e];
    int d = dst[e];
    const float4 v = *(const float4*)(h + (size_t)s * 128 + lane * 4);
    float* p = acc + (size_t)d * 128 + lane * 4;
    atomic_add_f32_dev(p + 0, v.x);
    atomic_add_f32_dev(p + 1, v.y);
    atomic_add_f32_dev(p + 2, v.z);
    atomic_add_f32_dev(p + 3, v.w);
}

// Y[N, DOUT] = X[N, 128] @ W[128, DOUT] using V_WMMA_F32_16X16X4_F32.
// One wave computes one 16x16 tile; K iterated in steps of 4 (32 WMMAs).
// VGPR layouts per CDNA5 ISA 7.12.2:
//   A 16x4 : lanes 0-15 M=0..15 K={0,1}; lanes 16-31 K={2,3}
//   B 4x16 : lanes 0-15 N=0..15 K={0,1}; lanes 16-31 K={2,3}
//   C/D    : VGPR r -> M = r + 8*half, N = lane%16
template <int DOUT>
__global__ void gin_gemm_wmma_f32(const float* __restrict__ X,
                                  const float* __restrict__ W,
                                  float* __restrict__ Y) {
    const int lane = threadIdx.x & 31;
    const int wave = threadIdx.x >> 5;
    const int lid  = lane & 15;
    const int half = lane >> 4;

    const int tiles_n = DOUT / 16;
    const int tile_m  = blockIdx.x;
    const int tile_n  = blockIdx.y * 4 + wave;   // wave-uniform
    if (tile_n >= tiles_n) return;               // wave-uniform branch: EXEC all-1s inside

    const int m0 = tile_m * 16;
    const int n0 = tile_n * 16;

    const float* xrow = X + (size_t)(m0 + lid) * 128;
    const float* wcol = W + n0 + lid;

    v8f c = {};
#pragma unroll
    for (int k0 = 0; k0 < 128; k0 += 4) {
        v2f a, b;
        a.x = xrow[k0 + 2 * half + 0];
        a.y = xrow[k0 + 2 * half + 1];
        b.x = wcol[(size_t)(k0 + 2 * half + 0) * DOUT];
        b.y = wcol[(size_t)(k0 + 2 * half + 1) * DOUT];
        // (neg_a, A, neg_b, B, c_mod, C, reuse_a, reuse_b)
        c = __builtin_amdgcn_wmma_f32_16x16x4_f32(false, a, false, b,
                                                  (short)0, c, false, false);
    }

#pragma unroll
    for (int r = 0; r < 8; ++r) {
        Y[(size_t)(m0 + r + 8 * half) * DOUT + n0 + lid] = c[r];
    }
}

extern "C" void kernel_launch(void* const* d_in, const int* in_sizes, int n_in,
                              void* d_out, int out_size, void* d_ws, size_t ws_size,
                              hipStream_t stream) {
    const float* feat = (const float*)d_in[0];   // [N,128]
    const float* W0   = (const float*)d_in[1];   // [128,128]
    const float* W1   = (const float*)d_in[2];   // [128,128]
    const float* W2   = (const float*)d_in[3];   // [128,64]
    const int*   srcI = (const int*)d_in[4];     // [E]
    const int*   dstI = (const int*)d_in[5];     // [E]

    const int N = in_sizes[0] / 128;             // 10000 (multiple of 16)
    const int E = in_sizes[4];                   // 640000

    float* t = (float*)d_ws;                     // [N,128] scratch: h + agg
    float* h = t + (size_t)N * 128;              // [N,128] scratch: activations
    float* out = (float*)d_out;                  // [N,64]

    const int n4 = N * 128 / 4;
    const int copyGrid = (n4 + 255) / 256;
    const long scatterThreads = (long)E * 32;
    const int scatterGrid = (int)((scatterThreads + 255) / 256);

    // ---- Layer 0: t = feat + agg(feat); h = t @ W0 (128->128) ----
    gin_copy_f32x4<<<copyGrid, 256, 0, stream>>>((const float4*)feat, (float4*)t, n4);
    gin_scatter_add<<<scatterGrid, 256, 0, stream>>>(feat, srcI, dstI, t, E);
    {
        dim3 grid(N / 16, 2);                    // 8 n-tiles / 4 waves
        gin_gemm_wmma_f32<128><<<grid, 128, 0, stream>>>(t, W0, h);
    }

    // ---- Layer 1: t = h + agg(h); h = t @ W1 (128->128) ----
    gin_copy_f32x4<<<copyGrid, 256, 0, stream>>>((const float4*)h, (float4*)t, n4);
    gin_scatter_add<<<scatterGrid, 256, 0, stream>>>(h, srcI, dstI, t, E);
    {
        dim3 grid(N / 16, 2);
        gin_gemm_wmma_f32<128><<<grid, 128, 0, stream>>>(t, W1, h);
    }

    // ---- Layer 2: t = h + agg(h); out = t @ W2 (128->64) ----
    gin_copy_f32x4<<<copyGrid, 256, 0, stream>>>((const float4*)h, (float4*)t, n4);
    gin_scatter_add<<<scatterGrid, 256, 0, stream>>>(h, srcI, dstI, t, E);
    {
        dim3 grid(N / 16, 1);                    // 4 n-tiles / 4 waves
        gin_gemm_wmma_f32<64><<<grid, 128, 0, stream>>>(t, W2, out);
    }
}